// FACoef_36558761624428
// MI455X (gfx1250) — compile-verified
//
#include <hip/hip_runtime.h>

// FACoef on MI455X (gfx1250, wave32, WMMA).
// Chain: P1 = X@X, P2 = P1@X, P3 = P2@X  (64 batches of 512x512 fp32),
// fused per-step reductions sum(p), sum(p^2), sum(p^3), tiny finalize.
// Uses V_WMMA_F32_16X16X4_F32 for exact fp32 math (reference is fp32).

typedef float v2f __attribute__((ext_vector_type(2)));
typedef float v8f __attribute__((ext_vector_type(8)));

#define MAT_N   512
#define BATCH_B 64

// One block = 128 threads = 4 waves; block computes a 128x128 output tile.
// Each wave computes 64x64 = 4x4 WMMA tiles of 16x16, K swept in steps of 4.
__global__ __launch_bounds__(128)
void gemm_pow_trace(const float* __restrict__ A,   // (B,N,N) left operand
                    const float* __restrict__ X,   // (B,N,N) right operand (always x)
                    float* __restrict__ P,         // (B,N,N) product out
                    float* __restrict__ partial)   // (B,16,3) per-block partial sums
{
    const int b     = blockIdx.z;
    const int lane  = threadIdx.x & 31;
    const int wave  = threadIdx.x >> 5;
    const int laneM = lane & 15;   // M (or N) index within tile
    const int laneH = lane >> 4;   // which K-pair half (ISA 16x4 A layout)

    const float* __restrict__ Ab = A + (size_t)b * MAT_N * MAT_N;
    const float* __restrict__ Xb = X + (size_t)b * MAT_N * MAT_N;
    float* __restrict__       Pb = P + (size_t)b * MAT_N * MAT_N;

    const int row0 = blockIdx.y * 128 + (wave >> 1) * 64;  // wave's output origin
    const int col0 = blockIdx.x * 128 + (wave & 1) * 64;

    v8f acc[4][4];
    #pragma unroll
    for (int mt = 0; mt < 4; ++mt)
        #pragma unroll
        for (int nt = 0; nt < 4; ++nt)
            acc[mt][nt] = (v8f){0.f, 0.f, 0.f, 0.f, 0.f, 0.f, 0.f, 0.f};

    for (int k = 0; k < MAT_N; k += 4) {
        // A fragments (16x4, fp32): lane L holds row M=L%16; VGPR pair holds
        // K = 2*(L/16) and K = 2*(L/16)+1  -> one b64 load per tile.
        v2f aF[4];
        #pragma unroll
        for (int mt = 0; mt < 4; ++mt) {
            const float* ap = Ab + (size_t)(row0 + mt * 16 + laneM) * MAT_N
                                 + (k + 2 * laneH);
            aF[mt] = *(const v2f*)ap;
        }
        // B fragments (4x16, fp32): VGPR0 = B[2*laneH][col], VGPR1 = B[2*laneH+1][col].
        // Lanes 0-15 read 64B contiguous, lanes 16-31 read 64B contiguous.
        v2f bF[4];
        #pragma unroll
        for (int nt = 0; nt < 4; ++nt) {
            const float* bp = Xb + (size_t)(k + 2 * laneH) * MAT_N
                                 + (col0 + nt * 16 + laneM);
            bF[nt].x = bp[0];
            bF[nt].y = bp[MAT_N];
        }
        // 16 WMMAs: 16*16*4*2 = 2048 FLOP each, full fp32.
        #pragma unroll
        for (int mt = 0; mt < 4; ++mt)
            #pragma unroll
            for (int nt = 0; nt < 4; ++nt)
                acc[mt][nt] = __builtin_amdgcn_wmma_f32_16x16x4_f32(
                    false, aF[mt], false, bF[nt],
                    (short)0, acc[mt][nt], false, false);
    }

    // Epilogue: store product, fuse sum(p), sum(p^2), sum(p^3) over this
    // wave's 64x64 region. C/D layout: VGPR r = row (r + 8*laneH), col = laneM.
    float s1 = 0.f, s2 = 0.f, s3 = 0.f;
    #pragma unroll
    for (int mt = 0; mt < 4; ++mt) {
        #pragma unroll
        for (int nt = 0; nt < 4; ++nt) {
            #pragma unroll
            for (int r = 0; r < 8; ++r) {
                float p = acc[mt][nt][r];
                int row = row0 + mt * 16 + r + 8 * laneH;
                int col = col0 + nt * 16 + laneM;
                Pb[(size_t)row * MAT_N + col] = p;
                float p2 = p * p;
                s1 += p;
                s2 += p2;
                s3 += p2 * p;
            }
        }
    }

    // Wave32 reduction.
    #pragma unroll
    for (int off = 16; off >= 1; off >>= 1) {
        s1 += __shfl_down(s1, off, 32);
        s2 += __shfl_down(s2, off, 32);
        s3 += __shfl_down(s3, off, 32);
    }

    __shared__ float red[4][3];
    if (lane == 0) {
        red[wave][0] = s1;
        red[wave][1] = s2;
        red[wave][2] = s3;
    }
    __syncthreads();
    if (threadIdx.x == 0) {
        float t0 = 0.f, t1 = 0.f, t2 = 0.f;
        #pragma unroll
        for (int w = 0; w < 4; ++w) {
            t0 += red[w][0];
            t1 += red[w][1];
            t2 += red[w][2];
        }
        int tile = blockIdx.y * 4 + blockIdx.x;              // 0..15 within batch
        float* dst = partial + ((size_t)b * 16 + tile) * 3;  // written exactly once
        dst[0] = t0;
        dst[1] = t1;
        dst[2] = t2;
    }
}

// out[b] = sum_{i,j} coef[i,j] * traces[b,i,j] / (N*N)^(i+j+2)
// (N*N)^(i+j+2) = 2^(18*(i+j+2)) exactly.
__global__ void finalize_traces(const float* __restrict__ coef,
                                const float* __restrict__ partial,  // (3,B,16,3)
                                float* __restrict__ out)
{
    int b = threadIdx.x;
    if (b >= BATCH_B) return;
    float r = 0.f;
    #pragma unroll
    for (int i = 0; i < 3; ++i) {
        float s[3] = {0.f, 0.f, 0.f};
        for (int t = 0; t < 16; ++t) {
            const float* p = partial
                + ((size_t)i * BATCH_B * 16 + (size_t)b * 16 + t) * 3;
            s[0] += p[0];
            s[1] += p[1];
            s[2] += p[2];
        }
        #pragma unroll
        for (int j = 0; j < 3; ++j) {
            float inv_norm = exp2f(-18.0f * (float)(i + j + 2));
            r += coef[i * 3 + j] * s[j] * inv_norm;
        }
    }
    out[b] = r;
}

extern "C" void kernel_launch(void* const* d_in, const int* in_sizes, int n_in,
                              void* d_out, int out_size, void* d_ws, size_t ws_size,
                              hipStream_t stream) {
    const float* x    = (const float*)d_in[0];  // (64,512,512) fp32
    const float* coef = (const float*)d_in[1];  // (3,3) fp32
    float*       out  = (float*)d_out;          // (64,) fp32

    // Workspace layout: ping-pong product buffers + partial-sum area.
    const size_t matElems = (size_t)BATCH_B * MAT_N * MAT_N;  // 16,777,216 floats = 64MB
    float* bufA     = (float*)d_ws;
    float* bufB     = bufA + matElems;
    float* partials = bufB + matElems;  // 3 * 64 * 16 * 3 floats = 36KB
    (void)in_sizes; (void)n_in; (void)out_size; (void)ws_size;

    dim3 grid(4, 4, BATCH_B);  // 128x128 tiles over 512x512, per batch
    dim3 block(128);

    // x^2, x^3, x^4 with fused power-sum reductions per step.
    gemm_pow_trace<<<grid, block, 0, stream>>>(x,    x, bufA, partials);
    gemm_pow_trace<<<grid, block, 0, stream>>>(bufA, x, bufB, partials + (size_t)BATCH_B * 16 * 3);
    gemm_pow_trace<<<grid, block, 0, stream>>>(bufB, x, bufA, partials + (size_t)2 * BATCH_B * 16 * 3);

    finalize_traces<<<1, 64, 0, stream>>>(coef, partials, out);
}